// CGAPipeline_38190849196335
// MI455X (gfx1250) — compile-verified
//
#include <hip/hip_runtime.h>
#include <hip/hip_bf16.h>
#include <stdint.h>

namespace {

constexpr int TPB = 256;
constexpr int ROW = 20;              // dwords per versor row in LDS (16 data + 4 pad, 16B aligned)
constexpr int WROW = 32 * ROW;       // versor region per wave (dwords)
constexpr int WX = 96;               // x staging per wave (32 pts * 3 dwords)
constexpr int WO = 96;               // out staging per wave
constexpr int WSTRIDE = WROW + WX + WO;  // 832 dwords per wave (16B multiple)

// ---- Cl(4,1) Cayley tables, computed at compile time (fold to immediates) ----
struct Tables {
  signed char t1o[16][5];   // even*e_p -> odd blade index
  signed char t1s[16][5];   // sign
  signed char t2q[16][16];  // odd*~even -> grade-1 index, -1 if not grade-1
  signed char t2s[16][16];  // sign (includes reverse sign of even blade)
};

constexpr int popcnt5(int x) { int c = 0; while (x) { c += x & 1; x >>= 1; } return c; }

constexpr Tables make_tables() {
  Tables T{};
  int EVEN[16] = {}, ODD[16] = {};
  int ne = 0, no = 0;
  for (int b = 0; b < 32; ++b) { if ((popcnt5(b) & 1) == 0) EVEN[ne++] = b; else ODD[no++] = b; }
  int oidx[32], g1idx[32];
  for (int i = 0; i < 32; ++i) { oidx[i] = -1; g1idx[i] = -1; }
  for (int i = 0; i < 16; ++i) oidx[ODD[i]] = i;
  g1idx[1] = 0; g1idx[2] = 1; g1idx[4] = 2; g1idx[8] = 3; g1idx[16] = 4;

  for (int ie = 0; ie < 16; ++ie)
    for (int ip = 0; ip < 5; ++ip) {
      int a = EVEN[ie], b = 1 << ip;
      int s = 0;
      for (int t = a >> 1; t; t >>= 1) s += popcnt5(t & b);
      int sign = (s & 1) ? -1 : 1;
      if ((a & b) & 16) sign = -sign;          // e5 squares to -1
      T.t1o[ie][ip] = (signed char)oidx[a ^ b];
      T.t1s[ie][ip] = (signed char)sign;
    }
  for (int io = 0; io < 16; ++io)
    for (int ie = 0; ie < 16; ++ie) {
      int a = ODD[io], b = EVEN[ie];
      int s = 0;
      for (int t = a >> 1; t; t >>= 1) s += popcnt5(t & b);
      int sign = (s & 1) ? -1 : 1;
      if ((a & b) & 16) sign = -sign;          // metric
      int g = popcnt5(b);
      if ((g * (g - 1) / 2) & 1) sign = -sign; // reverse(~M) sign
      T.t2q[io][ie] = (signed char)g1idx[a ^ b];
      T.t2s[io][ie] = (signed char)sign;
    }
  return T;
}

constexpr Tables TAB = make_tables();

// Low 32 bits of a flat pointer into LDS == the LDS byte offset (ISA 10.2 aperture rule).
__device__ __forceinline__ unsigned lds_lo32(const void* p) {
  return (unsigned)(unsigned long long)(uintptr_t)p;
}

} // namespace

__global__ __launch_bounds__(TPB) void cga_sandwich_kernel(
    const float* __restrict__ versor, const float* __restrict__ x,
    float* __restrict__ out, int n) {
  __shared__ __align__(16) float sm[(TPB / 32) * WSTRIDE];

  const int tid  = threadIdx.x;
  const int lane = tid & 31;
  const int wave = tid >> 5;
  const long long wbase = (long long)blockIdx.x * TPB + (long long)wave * 32;
  float* wsm  = sm + wave * WSTRIDE;   // versor rows (padded)
  float* wx   = wsm + WROW;            // x staging
  float* wout = wx + WX;               // out staging

  // ---- async DMA #1: 32 versor rows (2 KB) -> LDS; 512B contiguous per instr ----
#pragma unroll
  for (int j = 0; j < 4; ++j) {
    const int c = j * 32 + lane;            // 16B chunk id within wave tile
    long long pt = wbase + (c >> 2);
    if (pt >= n) pt = (n > 0) ? (long long)n - 1 : 0;  // clamp tail (reads only)
    const float* gsrc = versor + pt * 16 + (long long)(c & 3) * 4;
    const unsigned ldst = lds_lo32(wsm + (c >> 2) * ROW + (c & 3) * 4);
    asm volatile("global_load_async_to_lds_b128 %0, %1, off"
                 :: "v"(ldst), "v"((unsigned long long)(uintptr_t)gsrc)
                 : "memory");
  }

  // ---- async DMA #2: 96 dwords of x (384B contiguous) -> LDS; 128B per instr ----
  const long long xdw_total = 3LL * n;
#pragma unroll
  for (int k = 0; k < 3; ++k) {
    long long d = wbase * 3 + k * 32 + lane;
    if (d >= xdw_total) d = (xdw_total > 0) ? xdw_total - 1 : 0;  // clamp tail
    const float* gsrc = x + d;
    const unsigned ldst = lds_lo32(wx + k * 32 + lane);
    asm volatile("global_load_async_to_lds_b32 %0, %1, off"
                 :: "v"(ldst), "v"((unsigned long long)(uintptr_t)gsrc)
                 : "memory");
  }

  asm volatile("s_wait_asynccnt 0x0" ::: "memory");

  // Own versor row from LDS (16B-aligned -> ds_load_b128 x4).
  float v[16];
  {
    const float4* vr = reinterpret_cast<const float4*>(wsm + lane * ROW);
#pragma unroll
    for (int k = 0; k < 4; ++k) {
      float4 t = vr[k];
      v[4 * k + 0] = t.x; v[4 * k + 1] = t.y; v[4 * k + 2] = t.z; v[4 * k + 3] = t.w;
    }
  }
  // Own x from LDS (3L+j addressing: bank-conflict-free).
  const float px = wx[3 * lane + 0];
  const float py = wx[3 * lane + 1];
  const float pz = wx[3 * lane + 2];

  // ---- CGA encode: P = (x, y, z, 0.5|x|^2 - 0.5, 0.5|x|^2 + 0.5) ----
  float pp[5];
  const float half_sq = 0.5f * (px * px + py * py + pz * pz);
  pp[0] = px; pp[1] = py; pp[2] = pz; pp[3] = half_sq - 0.5f; pp[4] = half_sq + 0.5f;

  // ---- Stage 1: mx = M * P  (80 sign-folded FMAs) ----
  float mx[16];
#pragma unroll
  for (int k = 0; k < 16; ++k) mx[k] = 0.f;
#pragma unroll
  for (int e = 0; e < 16; ++e) {
#pragma unroll
    for (int ip = 0; ip < 5; ++ip) {
      mx[(int)TAB.t1o[e][ip]] += (float)TAB.t1s[e][ip] * v[e] * pp[ip];
    }
  }

  // ---- Stage 2: q = grade1(mx * ~M)  (~80 sign-folded FMAs after DCE) ----
  float q[5] = {0.f, 0.f, 0.f, 0.f, 0.f};
#pragma unroll
  for (int io = 0; io < 16; ++io) {
#pragma unroll
    for (int e = 0; e < 16; ++e) {
      const int qq = (int)TAB.t2q[io][e];
      if (qq >= 0) q[qq] += (float)TAB.t2s[io][e] * mx[io] * v[e];
    }
  }

  // ---- Decode into LDS staging tile (valid lanes only) ----
  const long long i = wbase + lane;
  if (i < n) {
    const float s = q[4] - q[3];
    const float inv = 1.0f / s;
    wout[3 * lane + 0] = q[0] * inv;
    wout[3 * lane + 1] = q[1] * inv;
    wout[3 * lane + 2] = q[2] * inv;
  }

  // ---- async drain: LDS -> out, 128B contiguous per instr; EXEC-guarded tail ----
  asm volatile("s_wait_dscnt 0x0" ::: "memory");
#pragma unroll
  for (int k = 0; k < 3; ++k) {
    const long long d = wbase * 3 + k * 32 + lane;
    if (d < xdw_total) {   // out has exactly 3n dwords
      float* gdst = out + d;
      const unsigned lsrc = lds_lo32(wout + k * 32 + lane);
      asm volatile("global_store_async_from_lds_b32 %0, %1, off"
                   :: "v"((unsigned long long)(uintptr_t)gdst), "v"(lsrc)
                   : "memory");
    }
  }
  // s_endpgm performs an implicit wait-idle, covering outstanding ASYNCcnt stores.
}

extern "C" void kernel_launch(void* const* d_in, const int* in_sizes, int n_in,
                              void* d_out, int out_size, void* d_ws, size_t ws_size,
                              hipStream_t stream) {
  const float* versor = (const float*)d_in[0];
  const float* x      = (const float*)d_in[1];
  float* out = (float*)d_out;
  const int n = in_sizes[1] / 3;
  const int blocks = (n + TPB - 1) / TPB;
  hipLaunchKernelGGL(cga_sandwich_kernel, dim3(blocks), dim3(TPB), 0, stream,
                     versor, x, out, n);
}